// PEER_30442728194580
// MI455X (gfx1250) — compile-verified
//
#include <hip/hip_runtime.h>
#include <math.h>

typedef float v2f __attribute__((ext_vector_type(2)));
typedef float v8f __attribute__((ext_vector_type(8)));
typedef unsigned int u32x4 __attribute__((ext_vector_type(4)));
typedef int i32x4 __attribute__((ext_vector_type(4)));
typedef int i32x8 __attribute__((ext_vector_type(8)));

#define H 8
#define QD 256
#define HALF 128
#define NKEY 32
#define DTK 4
#define TOKENS 1024      // B*S
#define D 1024
#define NQ 2048          // H*QD
#define OUTD 1024
#define NITEM 8192       // TOKENS*H

// GEMM tiling
#define BM 64            // rows per block
#define BN 128           // cols per block
#define BK 32            // K-step
#define BNP 129          // padded B row stride (via TDM pad: 1 dword per 128)
#define AMP 33           // padded A row stride

// ---------------- helpers ----------------
__device__ inline float waveRedSum32(float x) {
    #pragma unroll
    for (int off = 16; off; off >>= 1) x += __shfl_xor(x, off, 32);
    return x;
}

// Issue a TDM load of a 32x128 f32 tile of Wq (row stride 2048 elems) into LDS
// at byte offset lds_off, padded +1 dword per 128 dwords (-> 129-dword rows).
__device__ inline void tdm_load_B(const float* gaddr, unsigned int lds_off) {
    u32x4 g0;
    g0.x = 1u;                                   // count=1, user mode
    g0.y = lds_off;                              // LDS byte address
    unsigned long long ga = (unsigned long long)(uintptr_t)gaddr;
    g0.z = (unsigned int)ga;                     // global_addr[31:0]
    g0.w = (unsigned int)(ga >> 32) | 0x80000000u; // global_addr[56:32] | type=2
    i32x8 g1;
    g1[0] = (int)((2u << 16)       // data_size = 4 bytes
                | (1u << 20)       // pad_enable
                | (6u << 22));     // pad_interval = 128 dwords (pad_amount=0 -> 1 dword)
    g1[1] = (int)(2048u << 16);    // tensor_dim0 = 2048 (elements per row)
    g1[2] = (int)(1024u << 16);    // tensor_dim1 = 1024 rows (dim0 hi bits = 0)
    g1[3] = (int)(128u << 16);     // tile_dim0 = 128 (dim1 hi bits = 0)
    g1[4] = 32;                    // tile_dim1 = 32, tile_dim2 = 0
    g1[5] = 2048;                  // tensor_dim0_stride[31:0] = 2048
    g1[6] = 0;                     // stride hi / dim1_stride lo
    g1[7] = 0;
    i32x4 gz4 = {0, 0, 0, 0};              // groups 2/3 unused (2D tensor)
    i32x8 gz8 = {0, 0, 0, 0, 0, 0, 0, 0};
    __builtin_amdgcn_tensor_load_to_lds(g0, g1, gz4, gz4, gz8, 0);
}

// ---------------- kernel: normalize sub-keys ----------------
__global__ void peer_knorm(const float* __restrict__ k0,
                           const float* __restrict__ k1,
                           float* __restrict__ n0,
                           float* __restrict__ n1) {
    int wid = threadIdx.x >> 5, lane = threadIdx.x & 31;
    int gw = blockIdx.x * 8 + wid;          // [0,64)
    int table = gw >> 5, row = gw & 31;
    const float* src = table ? k1 : k0;
    float*       dst = table ? n1 : n0;
    float v[4]; float ss = 0.f;
    #pragma unroll
    for (int j = 0; j < 4; ++j) {
        v[j] = src[row * HALF + lane * 4 + j];
        ss += v[j] * v[j];
    }
    ss = waveRedSum32(ss);
    float inv = 1.f / fmaxf(sqrtf(ss), 1e-12f);
    #pragma unroll
    for (int j = 0; j < 4; ++j) dst[row * HALF + lane * 4 + j] = v[j] * inv;
}

// ---------------- kernel: Q projection GEMM (f32 WMMA + TDM double-buffer) ----------------
// q[TOKENS][NQ] = hs[TOKENS][D] @ Wq[D][NQ] + bq
// Block tile 64x128, 8 waves in 2x4 grid, each wave a 2x2 grid of 16x16 tiles.
// B tiles arrive via Tensor Data Mover (async, double buffered, padded rows).
__global__ void peer_qproj(const float* __restrict__ hs,
                           const float* __restrict__ Wq,
                           const float* __restrict__ bq,
                           float* __restrict__ q) {
    __shared__ float As[2][BM][AMP];
    __shared__ float Bs[2][BK][BNP];
    int t = threadIdx.x;
    int wid = t >> 5, lane = t & 31;
    int rowBase = blockIdx.y * BM;
    int colBase = blockIdx.x * BN;
    int wm = wid >> 2;                 // 0..1  (row group of 32)
    int wn = wid & 3;                  // 0..3  (col group of 32)
    int l15 = lane & 15;
    int khi = (lane >> 4) << 1;        // 0 or 2

    v8f acc00 = {}, acc01 = {}, acc10 = {}, acc11 = {};

    // ---- prologue: stage 0 ----
    if (wid == 0)
        tdm_load_B(Wq + colBase, (unsigned int)(uintptr_t)&Bs[0][0][0]);
    #pragma unroll
    for (int i = 0; i < 8; ++i) {                // A: 64x32
        int idx = i * 256 + t;
        int r = idx >> 5, c = idx & 31;
        As[0][r][c] = hs[(rowBase + r) * D + c];
    }

    int buf = 0;
    for (int it = 0; it < D / BK; ++it) {
        int k0 = it * BK;
        // ---- issue next stage (overlaps this stage's WMMA) ----
        if (it < D / BK - 1) {
            int kn = k0 + BK;
            if (wid == 0)
                tdm_load_B(Wq + (size_t)kn * NQ + colBase,
                           (unsigned int)(uintptr_t)&Bs[buf ^ 1][0][0]);
            #pragma unroll
            for (int i = 0; i < 8; ++i) {
                int idx = i * 256 + t;
                int r = idx >> 5, c = idx & 31;
                As[buf ^ 1][r][c] = hs[(rowBase + r) * D + kn + c];
            }
            if (wid == 0) __builtin_amdgcn_s_wait_tensorcnt(1);
        } else {
            if (wid == 0) __builtin_amdgcn_s_wait_tensorcnt(0);
        }
        __syncthreads();    // stage `buf` (A stores + TDM B) complete

        #pragma unroll
        for (int kk = 0; kk < 8; ++kk) {
            int kb = kk * 4 + khi;
            v2f a0 = { As[buf][wm * 32 + l15][kb],      As[buf][wm * 32 + l15][kb + 1] };
            v2f a1 = { As[buf][wm * 32 + 16 + l15][kb], As[buf][wm * 32 + 16 + l15][kb + 1] };
            v2f b0 = { Bs[buf][kb][wn * 32 + l15],      Bs[buf][kb + 1][wn * 32 + l15] };
            v2f b1 = { Bs[buf][kb][wn * 32 + 16 + l15], Bs[buf][kb + 1][wn * 32 + 16 + l15] };
            acc00 = __builtin_amdgcn_wmma_f32_16x16x4_f32(false, a0, false, b0, (short)0, acc00, false, false);
            acc01 = __builtin_amdgcn_wmma_f32_16x16x4_f32(false, a0, false, b1, (short)0, acc01, false, false);
            acc10 = __builtin_amdgcn_wmma_f32_16x16x4_f32(false, a1, false, b0, (short)0, acc10, false, false);
            acc11 = __builtin_amdgcn_wmma_f32_16x16x4_f32(false, a1, false, b1, (short)0, acc11, false, false);
        }
        __syncthreads();    // all waves done reading stage `buf`
        buf ^= 1;
    }

    // ---- epilogue: bias + store (C layout: VGPR v -> M = v + 8*(lane>=16)) ----
    int rhalf = (lane >> 4) << 3;
    #pragma unroll
    for (int sn = 0; sn < 2; ++sn) {
        int col = colBase + wn * 32 + sn * 16 + l15;
        float bias = bq[col];
        v8f a0v = sn ? acc01 : acc00;
        v8f a1v = sn ? acc11 : acc10;
        #pragma unroll
        for (int v = 0; v < 8; ++v) {
            int row0 = rowBase + wm * 32 + v + rhalf;
            q[row0 * NQ + col] = a0v[v] + bias;
            q[(row0 + 16) * NQ + col] = a1v[v] + bias;
        }
    }
}

// ---------------- kernel: BN partial stats ----------------
__global__ void peer_bnstat(const float* __restrict__ q,
                            float* __restrict__ part1,
                            float* __restrict__ part2) {
    int t = threadIdx.x;
    float s = 0.f, s2 = 0.f;
    int r0 = blockIdx.x * 32;
    for (int r = r0; r < r0 + 32; ++r) {
        #pragma unroll
        for (int i = 0; i < 8; ++i) {
            float v = q[r * NQ + i * QD + t];
            s += v; s2 += v * v;
        }
    }
    part1[blockIdx.x * QD + t] = s;
    part2[blockIdx.x * QD + t] = s2;
}

// ---------------- kernel: BN finalize ----------------
__global__ void peer_bnfin(const float* __restrict__ part1,
                           const float* __restrict__ part2,
                           const float* __restrict__ gamma,
                           const float* __restrict__ beta,
                           float* __restrict__ bnsc,
                           float* __restrict__ bnsh) {
    int t = threadIdx.x;
    float s = 0.f, s2 = 0.f;
    for (int b = 0; b < 32; ++b) {
        s  += part1[b * QD + t];
        s2 += part2[b * QD + t];
    }
    float mean = s * (1.f / 8192.f);
    float var  = s2 * (1.f / 8192.f) - mean * mean;
    float sc = gamma[t] / sqrtf(var + 1e-5f);
    bnsc[t] = sc;
    bnsh[t] = beta[t] - mean * sc;
}

// ---------------- kernel: routing ----------------
// one wave per (token, head); TOPK==DTK^2 so softmax runs over ALL 16 combos.
__global__ void peer_route(const float* __restrict__ q,
                           const float* __restrict__ bnsc,
                           const float* __restrict__ bnsh,
                           const float* __restrict__ n0,
                           const float* __restrict__ n1,
                           int* __restrict__ rid,
                           float* __restrict__ rw) {
    __shared__ float qn[8][QD];
    int wid = threadIdx.x >> 5, lane = threadIdx.x & 31;
    int item = blockIdx.x * 8 + wid;            // [0, 8192)
    int r = item >> 3, h = item & 7;
    const float* qrow = q + r * NQ + h * QD;

    float vr[8]; float ss = 0.f;
    int cb = lane * 8;
    #pragma unroll
    for (int j = 0; j < 8; ++j) {
        int c = cb + j;
        float x = qrow[c] * bnsc[c] + bnsh[c];
        vr[j] = x; ss += x * x;
    }
    float ssg = ss;
    #pragma unroll
    for (int off = 8; off; off >>= 1) ssg += __shfl_xor(ssg, off, 16);
    float ss0 = __shfl(ssg, 0, 32);
    float ss1 = __shfl(ssg, 16, 32);
    float inv = (lane < 16) ? 1.f / fmaxf(sqrtf(ss0), 1e-12f)
                            : 1.f / fmaxf(sqrtf(ss1), 1e-12f);
    #pragma unroll
    for (int j = 0; j < 8; ++j) qn[wid][cb + j] = vr[j] * inv;
    __syncthreads();

    float s0 = 0.f, s1 = 0.f;
    const float* k0r = n0 + lane * HALF;
    const float* k1r = n1 + lane * HALF;
    #pragma unroll 4
    for (int c = 0; c < HALF; ++c) {
        s0 += qn[wid][c] * k0r[c];
        s1 += qn[wid][HALF + c] * k1r[c];
    }

    float ts0[DTK], ts1[DTK]; int ti0[DTK], ti1[DTK];
    float mv = s0; int mi = lane;
    #pragma unroll
    for (int tI = 0; tI < DTK; ++tI) {
        float v = mv; int i = mi;
        #pragma unroll
        for (int off = 16; off; off >>= 1) {
            float ov = __shfl_xor(v, off, 32);
            int   oi = __shfl_xor(i, off, 32);
            if (ov > v || (ov == v && oi < i)) { v = ov; i = oi; }
        }
        ts0[tI] = v; ti0[tI] = i;
        if (lane == i) mv = -INFINITY;
    }
    mv = s1; mi = lane;
    #pragma unroll
    for (int tI = 0; tI < DTK; ++tI) {
        float v = mv; int i = mi;
        #pragma unroll
        for (int off = 16; off; off >>= 1) {
            float ov = __shfl_xor(v, off, 32);
            int   oi = __shfl_xor(i, off, 32);
            if (ov > v || (ov == v && oi < i)) { v = ov; i = oi; }
        }
        ts1[tI] = v; ti1[tI] = i;
        if (lane == i) mv = -INFINITY;
    }

    int ii = (lane >> 2) & 3, jj = lane & 3;
    float cs = ts0[ii] + ts1[jj];
    int  cid = ti0[ii] * NKEY + ti1[jj];
    float m = cs;
    #pragma unroll
    for (int off = 8; off; off >>= 1) m = fmaxf(m, __shfl_xor(m, off, 16));
    float e = expf(cs - m);
    float sum = e;
    #pragma unroll
    for (int off = 8; off; off >>= 1) sum += __shfl_xor(sum, off, 16);
    float w = e / sum;
    if (lane < 16) {
        rid[item * 16 + lane] = cid;
        rw [item * 16 + lane] = w;
    }
}

// ---------------- kernel: expert gather/MLP/accumulate ----------------
__global__ void peer_expert(const float* __restrict__ hs,
                            const float* __restrict__ down,
                            const float* __restrict__ up,
                            const int* __restrict__ rid,
                            const float* __restrict__ rw,
                            float* __restrict__ out) {
    __shared__ float hsr[D];
    __shared__ float geh[128];
    __shared__ int   eid[128];
    __shared__ float ew [128];
    int t = threadIdx.x;
    int r = blockIdx.x;
    #pragma unroll
    for (int i = 0; i < 4; ++i) hsr[t + 256 * i] = hs[r * D + t + 256 * i];
    if (t < 128) {
        eid[t] = rid[r * 128 + t];
        ew [t] = rw [r * 128 + t];
    }
    __syncthreads();

    int wid = t >> 5, lane = t & 31;
    for (int e = wid * 16; e < wid * 16 + 16; ++e) {
        const float* dr = down + (size_t)eid[e] * D;
        float acc = 0.f;
        #pragma unroll 8
        for (int i = 0; i < 32; ++i) acc += dr[i * 32 + lane] * hsr[i * 32 + lane];
        acc = waveRedSum32(acc);
        if (lane == 0) {
            float x = acc;
            float g = 0.5f * x * (1.0f + erff(x * 0.70710678118654752f)); // exact gelu
            geh[e] = g * ew[e];
        }
    }
    __syncthreads();

    float acc0 = 0.f, acc1 = 0.f, acc2 = 0.f, acc3 = 0.f;
    for (int e = 0; e < 128; ++e) {
        if (e + 1 < 128)    // hide L2 latency of the next gathered row
            __builtin_prefetch(up + (size_t)eid[e + 1] * OUTD + t, 0, 0);
        const float* ur = up + (size_t)eid[e] * OUTD;
        float g = geh[e];
        acc0 += g * ur[t];
        acc1 += g * ur[t + 256];
        acc2 += g * ur[t + 512];
        acc3 += g * ur[t + 768];
    }
    out[r * OUTD + t]       = acc0;
    out[r * OUTD + t + 256] = acc1;
    out[r * OUTD + t + 512] = acc2;
    out[r * OUTD + t + 768] = acc3;
}

// ---------------- launcher ----------------
extern "C" void kernel_launch(void* const* d_in, const int* in_sizes, int n_in,
                              void* d_out, int out_size, void* d_ws, size_t ws_size,
                              hipStream_t stream) {
    const float* hs    = (const float*)d_in[0];  // [1024,1024]
    const float* Wq    = (const float*)d_in[1];  // [1024,2048]
    const float* bq    = (const float*)d_in[2];  // [2048]
    const float* gamma = (const float*)d_in[3];  // [256]
    const float* beta  = (const float*)d_in[4];  // [256]
    const float* sk0   = (const float*)d_in[5];  // [32,128]
    const float* sk1   = (const float*)d_in[6];  // [32,128]
    const float* down  = (const float*)d_in[7];  // [1024,1024]
    const float* up    = (const float*)d_in[8];  // [1024,1024]
    float* out = (float*)d_out;                  // [1024,1024]

    float* ws = (float*)d_ws;
    float* q      = ws;                   // 1024*2048
    float* n0     = q      + TOKENS * NQ;
    float* n1     = n0     + NKEY * HALF;
    float* part1  = n1     + NKEY * HALF;
    float* part2  = part1  + 32 * QD;
    float* bnsc   = part2  + 32 * QD;
    float* bnsh   = bnsc   + QD;
    int*   rid    = (int*)(bnsh + QD);
    float* rw     = (float*)(rid + NITEM * 16);

    peer_knorm<<<dim3(8), dim3(256), 0, stream>>>(sk0, sk1, n0, n1);
    peer_qproj<<<dim3(NQ / BN, TOKENS / BM), dim3(256), 0, stream>>>(hs, Wq, bq, q);
    peer_bnstat<<<dim3(32), dim3(256), 0, stream>>>(q, part1, part2);
    peer_bnfin<<<dim3(1), dim3(256), 0, stream>>>(part1, part2, gamma, beta, bnsc, bnsh);
    peer_route<<<dim3(TOKENS), dim3(256), 0, stream>>>(q, bnsc, bnsh, n0, n1, rid, rw);
    peer_expert<<<dim3(TOKENS), dim3(256), 0, stream>>>(hs, down, up, rid, rw, out);
}